// ELModel_46729244180678
// MI455X (gfx1250) — compile-verified
//
#include <hip/hip_runtime.h>
#include <math.h>

// Problem constants (match reference)
#define D_EMB   51
#define DD_EMB  (D_EMB * D_EMB)       // 2601
#define B_BATCH 8192
#define MARGIN  0.0f
#define INF_R   5.0f

// Workspace partial-sum layout (floats)
#define NPAIR     (4 * B_BATCH)
#define OFF_PAIR  0
#define OFF_CHAIN (OFF_PAIR + NPAIR)            // 32768
#define OFF_NF2   (OFF_CHAIN + B_BATCH)         // 40960
#define OFF_TOP   (OFF_NF2 + 32)                // 40992
#define OFF_RAD   (OFF_TOP + 32)                // 41024
#define NTOT      (OFF_RAD + 32)                // 41056 floats ~ 164 KB

typedef __attribute__((ext_vector_type(2))) float v2f;
typedef __attribute__((ext_vector_type(8))) float v8f;

__device__ __forceinline__ float relu_f(float x) { return fmaxf(x, 0.0f); }

// LDS byte offset of a __shared__ object: the LDS aperture of a generic
// pointer lives entirely in the upper 32 bits, so the low 32 bits are the
// wave-relative LDS byte address (what async-to-LDS VDST expects).
__device__ __forceinline__ unsigned lds_off(const void* p) {
    return (unsigned)(uintptr_t)p;
}

// ---------------------------------------------------------------------------
// Pairwise losses: nf1 (mode 0), nf3 (mode 1), nf4 (mode 2), nf3_neg (mode 3)
// One 64-thread block (2 waves) per batch row. Threads 0..50 each compute one
// row of the two mat-vec products rm@c, rm@d (bandwidth-bound gather of the
// 10.4KB relation matrix), then a 3-way block reduction for the norms.
// ---------------------------------------------------------------------------
__global__ void __launch_bounds__(64)
pair_kernel(const float* __restrict__ cls, const float* __restrict__ rel,
            const int* __restrict__ idx, float* __restrict__ partial, int mode)
{
    __shared__ float cs[D_EMB], ds[D_EMB], crs[D_EMB], drs[D_EMB];
    __shared__ float r0[64], r1[64], r2[64];

    int b  = blockIdx.x;
    int t  = threadIdx.x;
    int i0 = idx[b * 3 + 0], i1 = idx[b * 3 + 1], i2 = idx[b * 3 + 2];
    int ci, di, ri;
    if (mode == 2) { ri = i0; ci = i1; di = i2; }  // nf4: rel idx first
    else           { ci = i0; ri = i1; di = i2; }

    const float* c  = cls + (size_t)ci * D_EMB;
    const float* d  = cls + (size_t)di * D_EMB;
    const float* rm = rel + (size_t)ri * DD_EMB;

    if (t < D_EMB) { cs[t] = c[t]; ds[t] = d[t]; }
    __syncthreads();

    if (t < D_EMB) {
        const float* row = rm + (size_t)t * D_EMB;
        float ac = 0.0f, ad = 0.0f;
        #pragma unroll 1
        for (int j = 0; j < D_EMB; ++j) {
            float r = row[j];
            ac = fmaf(r, cs[j], ac);
            ad = fmaf(r, ds[j], ad);
        }
        crs[t] = ac; drs[t] = ad;
    }
    __syncthreads();

    float pe = 0.0f, pc = 0.0f, pd = 0.0f;
    if (t < D_EMB - 1) {                // first 50 components only
        float diff = drs[t] - crs[t];
        pe = diff * diff;
        pc = cs[t] * cs[t];
        pd = ds[t] * ds[t];
    }
    r0[t] = pe; r1[t] = pc; r2[t] = pd;
    __syncthreads();
    for (int s = 32; s > 0; s >>= 1) {
        if (t < s) { r0[t] += r0[t + s]; r1[t] += r1[t + s]; r2[t] += r2[t + s]; }
        __syncthreads();
    }

    if (t == 0) {
        float euc  = sqrtf(r0[0]);
        float regc = fabsf(sqrtf(r1[0]) - 1.0f);
        float regd = fabsf(sqrtf(r2[0]) - 1.0f);
        float rc, rd;
        if (mode == 0) { rc = relu_f(crs[D_EMB - 1]); rd = relu_f(drs[D_EMB - 1]); } // nf1: transformed
        else           { rc = relu_f(cs[D_EMB - 1]);  rd = relu_f(ds[D_EMB - 1]);  }
        float dst;
        if      (mode == 0) dst = relu_f(euc + rc - rd + MARGIN);
        else if (mode == 1) dst = relu_f(euc + rc - rd - MARGIN);
        else if (mode == 2) dst = relu_f(euc - (rc + rd) - MARGIN);
        else                dst = -(euc - rc - rd) + MARGIN;     // nf3_neg, no relu
        partial[b] = dst + regc + regd;
    }
}

// ---------------------------------------------------------------------------
// Chain loss: per batch row, CD = C(51x51) @ Dm(51x51) in f32 via
// V_WMMA_F32_16X16X4_F32 on 64x64 zero-padded LDS tiles (stride 65 ->
// conflict-free banks), then mean |E - CD| over the real 51x51 window.
// Block = 128 threads (4 waves); wave w owns output rows [16w, 16w+16).
// Staging uses GLOBAL_LOAD_ASYNC_TO_LDS_B32 (ASYNCcnt path, no VGPR
// round-trip); E is prefetched (global_prefetch_b8) while the DMA fills LDS.
// ---------------------------------------------------------------------------
#define PADDIM 64
#define LDSTR  65

__global__ void __launch_bounds__(128)
chain_kernel(const float* __restrict__ rel, const int* __restrict__ idx,
             float* __restrict__ partial)
{
    __shared__ float Cs[PADDIM * LDSTR];
    __shared__ float Ds[PADDIM * LDSTR];
    __shared__ float red[128];

    int b   = blockIdx.x;
    int tid = threadIdx.x;
    const float* C  = rel + (size_t)idx[b * 3 + 0] * DD_EMB;
    const float* Dm = rel + (size_t)idx[b * 3 + 1] * DD_EMB;
    const float* E  = rel + (size_t)idx[b * 3 + 2] * DD_EMB;

    // Zero-fill padding (rows/cols >= 51 stay zero).
    for (int i = tid; i < PADDIM * LDSTR; i += 128) { Cs[i] = 0.0f; Ds[i] = 0.0f; }
    __syncthreads();   // ds stores complete before async writes can land

    // Async global -> LDS staging of both 51x51 matrices (B32: rel rows are
    // only 4B-aligned). SADDR form: vdst = LDS byte addr, vaddr = byte offset.
    for (int i = tid; i < DD_EMB; i += 128) {
        int r = i / D_EMB, cc = i - r * D_EMB;
        unsigned goff = (unsigned)i * 4u;
        unsigned lC = lds_off(&Cs[r * LDSTR + cc]);
        unsigned lD = lds_off(&Ds[r * LDSTR + cc]);
        asm volatile("global_load_async_to_lds_b32 %0, %1, %2"
                     :: "v"(lC), "v"(goff), "s"(C) : "memory");
        asm volatile("global_load_async_to_lds_b32 %0, %1, %2"
                     :: "v"(lD), "v"(goff), "s"(Dm) : "memory");
    }

    // Prefetch E (one 128B line per thread covers all 2601 floats) while the
    // async DMA fills LDS.
    {
        int pf = tid * 32;
        if (pf < DD_EMB) __builtin_prefetch(&E[pf], 0, 0);
    }

    asm volatile("s_wait_asynccnt 0x0" ::: "memory");
    __syncthreads();

    int wave  = tid >> 5;        // 0..3 -> output tile row
    int lane  = tid & 31;
    int lhalf = lane >> 4;       // 0: lanes 0-15, 1: lanes 16-31
    int l16   = lane & 15;
    int mbase = wave * 16;

    v8f acc[4] = {};
    // K padded to 52 (multiple of 4); fully unrolled (13 x 4 = 52 WMMAs) so
    // accumulators stay pinned in place (no inter-iteration copies).
    #pragma unroll
    for (int k0 = 0; k0 < 52; k0 += 4) {
        // A 16x4 layout: lanes 0-15 carry (K=k0, K=k0+1), lanes 16-31 (K=k0+2, K=k0+3)
        v2f a;
        int arow = mbase + l16;
        int acol = k0 + 2 * lhalf;
        a.x = Cs[arow * LDSTR + acol];
        a.y = Cs[arow * LDSTR + acol + 1];
        #pragma unroll
        for (int n = 0; n < 4; ++n) {
            // B 4x16 layout mirrors A: per-lane pair of consecutive K rows, N = l16
            v2f bf;
            int bcol = n * 16 + l16;
            int brow = k0 + 2 * lhalf;
            bf.x = Ds[brow * LDSTR + bcol];
            bf.y = Ds[(brow + 1) * LDSTR + bcol];
            acc[n] = __builtin_amdgcn_wmma_f32_16x16x4_f32(
                false, a, false, bf, (short)0, acc[n], false, false);
        }
    }

    // |E - CD| over valid 51x51 window. C/D layout: VGPR r, lanes 0-15 -> M=r,
    // lanes 16-31 -> M=r+8, N = l16 (+16n).
    float s = 0.0f;
    #pragma unroll
    for (int n = 0; n < 4; ++n) {
        #pragma unroll
        for (int r = 0; r < 8; ++r) {
            int M = mbase + r + 8 * lhalf;
            int N = n * 16 + l16;
            if (M < D_EMB && N < D_EMB)
                s += fabsf(E[M * D_EMB + N] - acc[n][r]);
        }
    }

    red[tid] = s;
    __syncthreads();
    for (int st = 64; st > 0; st >>= 1) {
        if (tid < st) red[tid] += red[tid + st];
        __syncthreads();
    }
    if (tid == 0) partial[b] = red[0] * (1.0f / (float)DD_EMB);
}

// ---------------------------------------------------------------------------
// nf2 / top / radius: thread-per-batch-row, block partial sums.
// ---------------------------------------------------------------------------
__global__ void __launch_bounds__(256)
nf2_kernel(const float* __restrict__ cls, const int* __restrict__ idx,
           float* __restrict__ partial)
{
    __shared__ float red[256];
    int b = blockIdx.x * 256 + threadIdx.x;
    float v = 0.0f;
    if (b < B_BATCH) {
        const float* c = cls + (size_t)idx[b * 3 + 0] * D_EMB;
        const float* d = cls + (size_t)idx[b * 3 + 1] * D_EMB;
        const float* e = cls + (size_t)idx[b * 3 + 2] * D_EMB;
        float s12 = 0, s13 = 0, s23 = 0, n1 = 0, n2 = 0, n3 = 0;
        for (int j = 0; j < D_EMB - 1; ++j) {
            float x1 = c[j], x2 = d[j], x3 = e[j];
            float a = x2 - x1, bb = x3 - x1, cc = x3 - x2;
            s12 = fmaf(a, a, s12); s13 = fmaf(bb, bb, s13); s23 = fmaf(cc, cc, s23);
            n1 = fmaf(x1, x1, n1); n2 = fmaf(x2, x2, n2); n3 = fmaf(x3, x3, n3);
        }
        float rc = relu_f(c[D_EMB - 1]), rd = relu_f(d[D_EMB - 1]), re = relu_f(e[D_EMB - 1]);
        float sr = rc + rd;
        float rdst = relu_f(fminf(rc, rd) - re - MARGIN);
        v = relu_f(sqrtf(s12) - sr - MARGIN)
          + relu_f(sqrtf(s13) - rc - MARGIN)
          + relu_f(sqrtf(s23) - rd - MARGIN)
          + rdst
          + fabsf(sqrtf(n1) - 1.0f) + fabsf(sqrtf(n2) - 1.0f) + fabsf(sqrtf(n3) - 1.0f);
    }
    red[threadIdx.x] = v;
    __syncthreads();
    for (int st = 128; st > 0; st >>= 1) {
        if (threadIdx.x < st) red[threadIdx.x] += red[threadIdx.x + st];
        __syncthreads();
    }
    if (threadIdx.x == 0) partial[blockIdx.x] = red[0];
}

__global__ void __launch_bounds__(256)
top_kernel(const float* __restrict__ cls, const int* __restrict__ idx,
           float* __restrict__ partial)
{
    __shared__ float red[256];
    int b = blockIdx.x * 256 + threadIdx.x;
    float v = 0.0f;
    if (b < B_BATCH) {
        float rd = relu_f(cls[(size_t)idx[b] * D_EMB + (D_EMB - 1)]);
        v = fabsf(rd - INF_R);
    }
    red[threadIdx.x] = v;
    __syncthreads();
    for (int st = 128; st > 0; st >>= 1) {
        if (threadIdx.x < st) red[threadIdx.x] += red[threadIdx.x + st];
        __syncthreads();
    }
    if (threadIdx.x == 0) partial[blockIdx.x] = red[0];
}

// radius partials are stored pre-negated: total loss uses (loss - loss7)
__global__ void __launch_bounds__(256)
radius_kernel(const float* __restrict__ cls, const int* __restrict__ idx,
              float* __restrict__ partial)
{
    __shared__ float red[256];
    int b = blockIdx.x * 256 + threadIdx.x;
    float v = 0.0f;
    if (b < B_BATCH) {
        float rd = cls[(size_t)idx[b] * D_EMB + (D_EMB - 1)];
        v = -fminf(0.0f, rd);
    }
    red[threadIdx.x] = v;
    __syncthreads();
    for (int st = 128; st > 0; st >>= 1) {
        if (threadIdx.x < st) red[threadIdx.x] += red[threadIdx.x + st];
        __syncthreads();
    }
    if (threadIdx.x == 0) partial[blockIdx.x] = red[0];
}

// ---------------------------------------------------------------------------
// Deterministic final reduction: single block sums the fixed-size partial
// array in a fixed order and writes mean over B.
// ---------------------------------------------------------------------------
__global__ void __launch_bounds__(256)
final_kernel(const float* __restrict__ partial, float* __restrict__ out)
{
    __shared__ float red[256];
    float s = 0.0f;
    for (int i = threadIdx.x; i < NTOT; i += 256) s += partial[i];
    red[threadIdx.x] = s;
    __syncthreads();
    for (int st = 128; st > 0; st >>= 1) {
        if (threadIdx.x < st) red[threadIdx.x] += red[threadIdx.x + st];
        __syncthreads();
    }
    if (threadIdx.x == 0) out[0] = red[0] * (1.0f / (float)B_BATCH);
}

// ---------------------------------------------------------------------------
extern "C" void kernel_launch(void* const* d_in, const int* in_sizes, int n_in,
                              void* d_out, int out_size, void* d_ws, size_t ws_size,
                              hipStream_t stream)
{
    (void)in_sizes; (void)n_in; (void)out_size; (void)ws_size;

    const int*   nf1    = (const int*)d_in[0];
    const int*   nf2i   = (const int*)d_in[1];
    const int*   nf3    = (const int*)d_in[2];
    const int*   nf4    = (const int*)d_in[3];
    const int*   top    = (const int*)d_in[4];
    const int*   nf3n   = (const int*)d_in[5];
    /* d_in[6] = nf_inclusion: unused by reference */
    const int*   chain  = (const int*)d_in[7];
    const int*   radius = (const int*)d_in[8];
    const float* cls    = (const float*)d_in[9];
    const float* rel    = (const float*)d_in[10];

    float* ws  = (float*)d_ws;
    float* out = (float*)d_out;

    pair_kernel<<<B_BATCH, 64, 0, stream>>>(cls, rel, nf1,  ws + OFF_PAIR + 0 * B_BATCH, 0);
    pair_kernel<<<B_BATCH, 64, 0, stream>>>(cls, rel, nf3,  ws + OFF_PAIR + 1 * B_BATCH, 1);
    pair_kernel<<<B_BATCH, 64, 0, stream>>>(cls, rel, nf4,  ws + OFF_PAIR + 2 * B_BATCH, 2);
    pair_kernel<<<B_BATCH, 64, 0, stream>>>(cls, rel, nf3n, ws + OFF_PAIR + 3 * B_BATCH, 3);

    chain_kernel<<<B_BATCH, 128, 0, stream>>>(rel, chain, ws + OFF_CHAIN);

    nf2_kernel<<<32, 256, 0, stream>>>(cls, nf2i,   ws + OFF_NF2);
    top_kernel<<<32, 256, 0, stream>>>(cls, top,    ws + OFF_TOP);
    radius_kernel<<<32, 256, 0, stream>>>(cls, radius, ws + OFF_RAD);

    final_kernel<<<1, 256, 0, stream>>>(ws, out);
}